// TtSSD_39367670235198
// MI455X (gfx1250) — compile-verified
//
#include <hip/hip_runtime.h>
#include <hip/hip_bf16.h>

// ---------------- constants ----------------
#define A_NUM      200000          // anchors
#define NCLS       21              // classes incl. background
#define NFG        20              // foreground classes
#define TOPK_K     400
#define NCAND      8192            // 20*400 = 8000, padded to pow2 for bitonic
#define MASK_WORDS (NCAND / 32)    // 256
#define SCORE_TH   0.01f
#define NMS_TH     0.45f
#define DETS       200
#define IMG_WH     320.0f
#define BBOX_CLAMP 4.135166556742356f   // log(1000/16)
#define CLS_OFF    1000.0f

typedef float v2f __attribute__((ext_vector_type(2)));
typedef float v8f __attribute__((ext_vector_type(8)));

#if __has_builtin(__builtin_amdgcn_global_load_async_to_lds_b128) && \
    __has_builtin(__builtin_amdgcn_s_wait_asynccnt)
#define USE_ASYNC_LDS 1
// exact pointee type from the compiler diagnostic: int __attribute__((vector_size(16)))
typedef int v4i_vec __attribute__((vector_size(4 * sizeof(int))));
typedef __attribute__((address_space(1))) v4i_vec* as1_v4i;
typedef __attribute__((address_space(3))) v4i_vec* as3_v4i;
#endif

// fast reciprocal (v_rcp_f32) -- validated-precision divide is wasted here
__device__ __forceinline__ float fast_rcp(float x) {
#if __has_builtin(__builtin_amdgcn_rcpf)
  return __builtin_amdgcn_rcpf(x);
#else
  return 1.0f / x;
#endif
}

// =====================================================================
// Kernel 1: softmax over 21 classes (row-sum via V_WMMA_F32_16X16X4_F32)
//           + box decode + clip.  One wave handles a 16-anchor tile.
// =====================================================================
__global__ void ssd_softmax_decode(const float* __restrict__ logits,
                                   const float* __restrict__ breg,
                                   const float* __restrict__ anchors,
                                   float* __restrict__ boxes,     // [A][4]
                                   float* __restrict__ scoresT) { // [20][A]
  const int wave = (blockIdx.x * blockDim.x + threadIdx.x) >> 5;
  const int lane = threadIdx.x & 31;
  const int tile = wave;                       // uniform per wave
  if (tile >= A_NUM / 16) return;              // wave-uniform exit (EXEC stays full)

  const int m    = lane & 15;                  // anchor row within tile
  const int half = lane >> 4;                  // K half: 0 -> k{0,1}, 1 -> k{2,3}
  const int a    = tile * 16 + m;

  const float* row = logits + (size_t)a * NCLS;

  // load this lane's 12 logits (classes padded to 24 with -1e30)
  float vals[12];
  float mx = -1e30f;
#pragma unroll
  for (int t = 0; t < 6; ++t) {
    const int c0 = 4 * t + 2 * half;
    const float v0 = (c0     < NCLS) ? row[c0]     : -1e30f;
    const float v1 = (c0 + 1 < NCLS) ? row[c0 + 1] : -1e30f;
    vals[2 * t] = v0; vals[2 * t + 1] = v1;
    mx = fmaxf(mx, fmaxf(v0, v1));
  }
  // row max across the lane pair (m, m+16)
  const float rmax = fmaxf(__shfl(mx, m), __shfl(mx, m + 16));

  float e[12];
#pragma unroll
  for (int t = 0; t < 12; ++t) e[t] = __expf(vals[t] - rmax);

  // WMMA: A(16x4) = exp tile chunk, B(4x16) = ones -> D[i][*] = rowsum(i)
  v2f bones; bones[0] = 1.0f; bones[1] = 1.0f;
  v8f acc = {};
#pragma unroll
  for (int t = 0; t < 6; ++t) {
    v2f av; av[0] = e[2 * t]; av[1] = e[2 * t + 1];
    acc = __builtin_amdgcn_wmma_f32_16x16x4_f32(
        /*neg_a=*/false, av, /*neg_b=*/false, bones,
        /*c_mod=*/(short)0, acc, /*reuse_a=*/false, /*reuse_b=*/false);
  }

  // D layout: VGPR r holds M=r (lanes 0-15) / M=r+8 (lanes 16-31).
  // rowsum(m): m<8 -> lane m, elt m ; m>=8 -> lane m+8, elt m-8.
  const int sel = m & 7;
  float elt = acc[0];
#pragma unroll
  for (int r = 1; r < 8; ++r) elt = (sel == r) ? acc[r] : elt;
  const int srcLane = (m < 8) ? m : (m + 8);
  const float inv = fast_rcp(__shfl(elt, srcLane));

  // write foreground probabilities, transposed [class-1][anchor]
#pragma unroll
  for (int t = 0; t < 6; ++t) {
    const int c0 = 4 * t + 2 * half;
    if (c0 >= 1 && c0 <= NFG)
      scoresT[(size_t)(c0 - 1) * A_NUM + a] = e[2 * t] * inv;
    const int c1 = c0 + 1;
    if (c1 >= 1 && c1 <= NFG)
      scoresT[(size_t)(c1 - 1) * A_NUM + a] = e[2 * t + 1] * inv;
  }

  // box decode + clip (one lane per anchor; half==0 lanes do it)
  if (half == 0) {
    const float4 rg = ((const float4*)breg)[a];
    const float4 an = ((const float4*)anchors)[a];
    const float aw = an.z - an.x, ah = an.w - an.y;
    const float acx = an.x + 0.5f * aw, acy = an.y + 0.5f * ah;
    const float dx = rg.x * 0.1f, dy = rg.y * 0.1f;
    const float dw = fminf(rg.z * 0.2f, BBOX_CLAMP);
    const float dh = fminf(rg.w * 0.2f, BBOX_CLAMP);
    const float pcx = dx * aw + acx, pcy = dy * ah + acy;
    const float pw = __expf(dw) * aw, ph = __expf(dh) * ah;
    float4 ob;
    ob.x = fminf(fmaxf(pcx - 0.5f * pw, 0.f), IMG_WH);
    ob.y = fminf(fmaxf(pcy - 0.5f * ph, 0.f), IMG_WH);
    ob.z = fminf(fmaxf(pcx + 0.5f * pw, 0.f), IMG_WH);
    ob.w = fminf(fmaxf(pcy + 0.5f * ph, 0.f), IMG_WH);
    ((float4*)boxes)[a] = ob;
  }
}

// =====================================================================
// Kernel 2: init candidate slots (invalid)
// =====================================================================
__global__ void ssd_init_cand(float* candScore, int* candAnchor, int* candLabel) {
  const int i = blockIdx.x * blockDim.x + threadIdx.x;
  if (i < NCAND) { candScore[i] = -1.0f; candAnchor[i] = 0; candLabel[i] = 0; }
}

// =====================================================================
// Kernel 3: per-class histogram top-400 select (one block per class)
// =====================================================================
__global__ void ssd_topk(const float* __restrict__ scoresT,
                         float* candScore, int* candAnchor, int* candLabel) {
  const int c = blockIdx.x;                    // 0..19
  const float* s = scoresT + (size_t)c * A_NUM;
  __shared__ int hist[1024];
  __shared__ int thrBin;
  __shared__ int counter;
  for (int b = threadIdx.x; b < 1024; b += blockDim.x) hist[b] = 0;
  if (threadIdx.x == 0) counter = 0;
  __syncthreads();

  for (int a = threadIdx.x; a < A_NUM; a += blockDim.x) {
    const float v = s[a];
    if (v > SCORE_TH) atomicAdd(&hist[min(1023, (int)(v * 1024.0f))], 1);
  }
  __syncthreads();
  if (threadIdx.x == 0) {
    int cum = 0, B = 0;
    for (int b = 1023; b >= 0; --b) { cum += hist[b]; if (cum >= TOPK_K) { B = b; break; } }
    thrBin = B;
  }
  __syncthreads();
  const float thr = (float)thrBin * (1.0f / 1024.0f);
  for (int a = threadIdx.x; a < A_NUM; a += blockDim.x) {
    const float v = s[a];
    if (v > SCORE_TH && v >= thr) {
      const int slot = atomicAdd(&counter, 1);
      if (slot < TOPK_K) {
        const int o = c * TOPK_K + slot;
        candScore[o] = v; candAnchor[o] = a; candLabel[o] = c + 1;
      }
    }
  }
}

// =====================================================================
// Kernel 4: single-block bitonic sort (score desc) + gather sorted arrays
//           8192 u64 keys = 64 KB LDS (fits CDNA5's 320 KB/WGP)
// =====================================================================
__device__ __forceinline__ unsigned fkey(float f) {
  unsigned u = __float_as_uint(f);
  return (u & 0x80000000u) ? ~u : (u | 0x80000000u);   // ascending-orderable
}

__global__ void __launch_bounds__(1024)
ssd_sort_gather(const float* __restrict__ candScore,
                const int* __restrict__ candAnchor,
                const int* __restrict__ candLabel,
                const float* __restrict__ boxes,
                float* sScore, int* sAnchor, int* sLabel,
                float* sBox, float* sNmsBox) {
  __shared__ unsigned long long keys[NCAND];
  const int tid = threadIdx.x;
  for (int i = tid; i < NCAND; i += 1024) {
    const unsigned inv = ~fkey(candScore[i]);          // ascending key = desc score
    keys[i] = ((unsigned long long)inv << 32) | (unsigned)i;
  }
  __syncthreads();
  for (int k = 2; k <= NCAND; k <<= 1) {
    for (int j = k >> 1; j > 0; j >>= 1) {
      for (int i = tid; i < NCAND; i += 1024) {
        const int ixj = i ^ j;
        if (ixj > i) {
          const unsigned long long x = keys[i], y = keys[ixj];
          const bool asc = ((i & k) == 0);
          if (asc ? (x > y) : (x < y)) { keys[i] = y; keys[ixj] = x; }
        }
      }
      __syncthreads();
    }
  }
  for (int i = tid; i < NCAND; i += 1024) {
    const int slot = (int)(keys[i] & 0xFFFFFFFFu);
    const float v = candScore[slot];
    const int an = candAnchor[slot];
    const int lb = candLabel[slot];
    sScore[i] = v; sAnchor[i] = an; sLabel[i] = lb;
    float4 bx = make_float4(0.f, 0.f, 0.f, 0.f);
    if (v > SCORE_TH) bx = ((const float4*)boxes)[an];
    ((float4*)sBox)[i] = bx;
    const float off = (float)lb * CLS_OFF;
    ((float4*)sNmsBox)[i] = make_float4(bx.x + off, bx.y + off, bx.z + off, bx.w + off);
  }
}

// =====================================================================
// Kernel 5: NMS suppression bitmask. 256 rows x 32 cols per block.
//           Column boxes staged in LDS via async global->LDS when available.
//           Division-free IoU test: inter/union > t  <=>  inter > t*union
//           (union >= 0; union==0 => inter==0 => false either way).
// =====================================================================
__global__ void ssd_nms_mask(const float* __restrict__ sNmsBox,
                             unsigned* __restrict__ mask) {
  __shared__ float colBox[32 * 4];
  __shared__ float colArea[32];
  const int tid = threadIdx.x;
  const int cb = blockIdx.y;                       // column block (32 boxes)
  if (tid < 32) {
    const float* src = sNmsBox + (size_t)(cb * 32 + tid) * 4;
#ifdef USE_ASYNC_LDS
    __builtin_amdgcn_global_load_async_to_lds_b128(
        (as1_v4i)(src), (as3_v4i)(&colBox[tid * 4]), 0, 0);
#else
    ((float4*)colBox)[tid] = ((const float4*)src)[0];
#endif
  }
#ifdef USE_ASYNC_LDS
  __builtin_amdgcn_s_wait_asynccnt(0);
#endif
  __syncthreads();
  if (tid < 32) {
    colArea[tid] = (colBox[tid * 4 + 2] - colBox[tid * 4 + 0]) *
                   (colBox[tid * 4 + 3] - colBox[tid * 4 + 1]);
  }
  __syncthreads();

  const int i = blockIdx.x * 256 + tid;            // row index
  const float4 rb = ((const float4*)sNmsBox)[i];
  const float areaR = (rb.z - rb.x) * (rb.w - rb.y);
  unsigned bits = 0u;
#pragma unroll
  for (int j = 0; j < 32; ++j) {
    const float x1 = fmaxf(rb.x, colBox[j * 4 + 0]);
    const float y1 = fmaxf(rb.y, colBox[j * 4 + 1]);
    const float x2 = fminf(rb.z, colBox[j * 4 + 2]);
    const float y2 = fminf(rb.w, colBox[j * 4 + 3]);
    const float w = fmaxf(x2 - x1, 0.f), h = fmaxf(y2 - y1, 0.f);
    const float inter = w * h;
    const float uni = areaR + colArea[j] - inter;
    const int col = cb * 32 + j;
    if ((inter > NMS_TH * uni) && col > i) bits |= (1u << j);
  }
  mask[(size_t)i * MASK_WORDS + cb] = bits;
}

// =====================================================================
// Kernel 6: single-block greedy bit-sweep + emit top-200 detections
// =====================================================================
__global__ void ssd_greedy_out(const float* __restrict__ sScore,
                               const int* __restrict__ sLabel,
                               const float* __restrict__ sBox,
                               const unsigned* __restrict__ mask,
                               float* __restrict__ out) {
  __shared__ unsigned removed[MASK_WORDS];
  __shared__ int keepList[DETS];
  __shared__ int nKeep;
  __shared__ int doSup;
  const int tid = threadIdx.x;                 // blockDim = 256 == MASK_WORDS
  removed[tid] = 0u;
  if (tid == 0) nKeep = 0;
  __syncthreads();

  for (int i = 0; i < NCAND; ++i) {
    if (tid == 0) {
      const bool alive = !((removed[i >> 5] >> (i & 31)) & 1u);
      const bool valid = sScore[i] > SCORE_TH;
      doSup = (alive && valid) ? 1 : 0;
      if (alive && valid && nKeep < DETS) keepList[nKeep++] = i;
    }
    __syncthreads();
    if (doSup) removed[tid] |= mask[(size_t)i * MASK_WORDS + tid];
    __syncthreads();
  }

  for (int k = tid; k < DETS; k += 256) {
    if (k < nKeep) {
      const int i = keepList[k];
      out[k * 4 + 0] = sBox[i * 4 + 0];
      out[k * 4 + 1] = sBox[i * 4 + 1];
      out[k * 4 + 2] = sBox[i * 4 + 2];
      out[k * 4 + 3] = sBox[i * 4 + 3];
      out[DETS * 4 + k] = sScore[i];
      out[DETS * 5 + k] = (float)sLabel[i];
    } else {
      out[k * 4 + 0] = 0.f; out[k * 4 + 1] = 0.f;
      out[k * 4 + 2] = 0.f; out[k * 4 + 3] = 0.f;
      out[DETS * 4 + k] = 0.f;
      out[DETS * 5 + k] = 0.f;
    }
  }
}

// =====================================================================
// Launcher
// =====================================================================
extern "C" void kernel_launch(void* const* d_in, const int* in_sizes, int n_in,
                              void* d_out, int out_size, void* d_ws, size_t ws_size,
                              hipStream_t stream) {
  const float* logits  = (const float*)d_in[0];   // [A,21]
  const float* breg    = (const float*)d_in[1];   // [A,4]
  const float* anchors = (const float*)d_in[2];   // [A,4]
  float* out = (float*)d_out;                     // 200*4 + 200 + 200 = 1200

  char* ws = (char*)d_ws;
  size_t off = 0;
  auto alloc = [&](size_t bytes) -> void* {
    void* p = ws + off;
    off = (off + bytes + 255) & ~(size_t)255;
    return p;
  };
  float* boxes      = (float*)alloc((size_t)A_NUM * 4 * sizeof(float));     // 3.2 MB
  float* scoresT    = (float*)alloc((size_t)NFG * A_NUM * sizeof(float));   // 16 MB
  float* candScore  = (float*)alloc(NCAND * sizeof(float));
  int*   candAnchor = (int*)  alloc(NCAND * sizeof(int));
  int*   candLabel  = (int*)  alloc(NCAND * sizeof(int));
  float* sScore     = (float*)alloc(NCAND * sizeof(float));
  int*   sAnchor    = (int*)  alloc(NCAND * sizeof(int));
  int*   sLabel     = (int*)  alloc(NCAND * sizeof(int));
  float* sBox       = (float*)alloc(NCAND * 4 * sizeof(float));
  float* sNmsBox    = (float*)alloc(NCAND * 4 * sizeof(float));
  unsigned* mask    = (unsigned*)alloc((size_t)NCAND * MASK_WORDS * sizeof(unsigned)); // 8 MB
  (void)ws_size; (void)in_sizes; (void)n_in; (void)out_size;

  // 12500 16-anchor tiles, 8 waves (256 thr) per block
  const int tiles = A_NUM / 16;
  const int blocks1 = (tiles + 7) / 8;
  ssd_softmax_decode<<<blocks1, 256, 0, stream>>>(logits, breg, anchors, boxes, scoresT);
  ssd_init_cand<<<NCAND / 256, 256, 0, stream>>>(candScore, candAnchor, candLabel);
  ssd_topk<<<NFG, 256, 0, stream>>>(scoresT, candScore, candAnchor, candLabel);
  ssd_sort_gather<<<1, 1024, 0, stream>>>(candScore, candAnchor, candLabel, boxes,
                                          sScore, sAnchor, sLabel, sBox, sNmsBox);
  ssd_nms_mask<<<dim3(NCAND / 256, MASK_WORDS), 256, 0, stream>>>(sNmsBox, mask);
  ssd_greedy_out<<<1, 256, 0, stream>>>(sScore, sLabel, sBox, mask, out);
}